// Visual_KNetNN_27616639713465
// MI455X (gfx1250) — compile-verified
//
#include <hip/hip_runtime.h>
#include <math.h>

// ---------------------------------------------------------------------------
// KalmanNet forward step for MI455X (gfx1250, wave32).
// Bandwidth-bound (~1 GB of fp32 FC2 weights per call @ 23.3 TB/s).
// Mid/large matvecs use V_WMMA_F32_16X16X4_F32: one wave per 16-row tile,
// x-vector placed in column 0 of the B operand, K-loop unrolled 8x with NT
// weight streaming. FC3/FC4 are dead code in the reference and are skipped.
// ---------------------------------------------------------------------------

typedef __attribute__((ext_vector_type(2))) float v2f;
typedef __attribute__((ext_vector_type(4))) float v4f;
typedef __attribute__((ext_vector_type(8))) float v8f;

#if defined(__HIP_DEVICE_COMPILE__)
#  if !__has_builtin(__builtin_amdgcn_wmma_f32_16x16x4_f32)
#    error "__builtin_amdgcn_wmma_f32_16x16x4_f32 not available for gfx1250 device pass"
#  endif
#endif

__device__ __forceinline__ v8f wmma_f32_16x16x4(v2f a, v2f b, v8f c)
{
#if defined(__HIP_DEVICE_COMPILE__)
    // 8 args: (neg_a, A, neg_b, B, c_mod, C, reuse_a, reuse_b)
    return __builtin_amdgcn_wmma_f32_16x16x4_f32(
        false, a, false, b, (short)0, c, false, false);
#else
    (void)a; (void)b;
    return c;   // host parse-only stub
#endif
}

// ---- workspace layout (float offsets) -------------------------------------
#define OFF_PRIOR   0      // m1x_prior            [32]
#define OFF_DY      32     // y - m1y              [32]
#define OFF_FC5     192    // out_FC5              [160]   (GRU_Q input)
#define OFF_Q       352    // out_Q                [1024]  (GRU_Sigma input starts here)
#define OFF_FC6     1376   // out_FC6              [160]   (contiguous: 1184-len concat)
#define OFF_FC1     1536   // out_FC1              [1024]  (GRU_S input starts here)
#define OFF_FC7     2560   // out_FC7              [320]   (contiguous: 1344-len concat)
#define OFF_SIGMA   2880   // out_Sigma            [1024]  (FC2 input starts here)
#define OFF_S       3904   // out_S                [1024]  (contiguous: 2048-len concat)
#define OFF_GI      4928   // GRU gi scratch       [3072]
#define OFF_GH      8000   // GRU gh scratch       [3072]
#define OFF_FC2OUT  11072  // out_FC2 (= KG flat)  [1024]
#define OFF_HIDDEN  12096  // FC2 hidden           [81920]

// ---------------------------------------------------------------------------
// Kernel 1: tiny prologue. m1x_prior, m1y, dy, 4 L2-normalized feature diffs,
// FC5 / FC6 / FC7 (all <= 320x64 — negligible work, one workgroup).
// ---------------------------------------------------------------------------
__global__ __launch_bounds__(128) void prep_kernel(
    const float* __restrict__ y, const float* __restrict__ post,
    const float* __restrict__ post_prev, const float* __restrict__ prior_prev,
    const float* __restrict__ y_prev, const float* __restrict__ F,
    const float* __restrict__ H,
    const float* __restrict__ FC5_w, const float* __restrict__ FC5_b,
    const float* __restrict__ FC6_w, const float* __restrict__ FC6_b,
    const float* __restrict__ FC7_w, const float* __restrict__ FC7_b,
    float* __restrict__ ws)
{
    __shared__ float s_prior[32], s_m1y[32], s_fe[32], s_fu[32], s_obs[64];
    __shared__ float s_inv[4];
    const int t = threadIdx.x;

    if (t < 32) {                                  // m1x_prior = F @ post
        float a = 0.f;
        for (int j = 0; j < 32; ++j) a += F[t * 32 + j] * post[j];
        s_prior[t] = a;
        ws[OFF_PRIOR + t] = a;
    }
    __syncthreads();
    if (t < 32) {                                  // m1y = H @ m1x_prior ; dy
        float a = 0.f;
        for (int j = 0; j < 32; ++j) a += H[t * 32 + j] * s_prior[j];
        s_m1y[t] = a;
        ws[OFF_DY + t] = y[t] - a;
    }
    __syncthreads();
    if (t < 4) {                                   // four L2 norms
        float ss = 0.f;
        for (int j = 0; j < 32; ++j) {
            float d;
            if      (t == 0) d = y[j] - y_prev[j];
            else if (t == 1) d = y[j] - s_m1y[j];
            else if (t == 2) d = post[j] - post_prev[j];
            else             d = post[j] - prior_prev[j];
            ss += d * d;
        }
        s_inv[t] = 1.0f / fmaxf(sqrtf(ss), 1e-12f);
    }
    __syncthreads();
    if (t < 32) {
        s_obs[t]      = (y[t] - y_prev[t])        * s_inv[0];
        s_obs[32 + t] = (y[t] - s_m1y[t])         * s_inv[1];
        s_fe[t]       = (post[t] - post_prev[t])  * s_inv[2];
        s_fu[t]       = (post[t] - prior_prev[t]) * s_inv[3];
    }
    __syncthreads();
    for (int r = t; r < 160; r += 128) {           // FC5 / FC6 (160x32)
        float a = FC5_b[r];
        for (int j = 0; j < 32; ++j) a += FC5_w[r * 32 + j] * s_fe[j];
        ws[OFF_FC5 + r] = fmaxf(a, 0.f);
        float c = FC6_b[r];
        for (int j = 0; j < 32; ++j) c += FC6_w[r * 32 + j] * s_fu[j];
        ws[OFF_FC6 + r] = fmaxf(c, 0.f);
    }
    for (int r = t; r < 320; r += 128) {           // FC7 (320x64)
        float a = FC7_b[r];
        for (int j = 0; j < 64; ++j) a += FC7_w[r * 64 + j] * s_obs[j];
        ws[OFF_FC7 + r] = fmaxf(a, 0.f);
    }
}

// ---------------------------------------------------------------------------
// WMMA dual GEMV: out = [relu](W @ x + b) for two independent (W,x,b,out)
// problems in one launch (pass tilesB=0 for a single matrix). One wave
// computes a 16-row tile via V_WMMA_F32_16X16X4_F32:
//   A tile : 16 rows x 4 K of W  (VGPR0 = K0|K2 halves, VGPR1 = K1|K3)
//   B tile : x[k0..k0+3] in column 0 only (lanes 0/16), rest zero
//   D col 0: lane 0 -> rows tb+0..7, lane 16 -> rows tb+8..15
// K-loop unrolled 8x (32 K per iteration): 8 NT b64 loads in flight, two
// alternating accumulators to break the WMMA D->C chain.
// Requires: rows % 16 == 0, K % 32 == 0, Ka+Kb <= 2432.
// ---------------------------------------------------------------------------
__global__ __launch_bounds__(256) void wmma_gemv2(
    const float* __restrict__ Wa, const float* __restrict__ xa,
    const float* __restrict__ ba, float* __restrict__ outa, int Ka, int tilesA,
    const float* __restrict__ Wb, const float* __restrict__ xb,
    const float* __restrict__ bb, float* __restrict__ outb, int Kb, int tilesB,
    int relu)
{
    __shared__ float ldsx[2432];
    const int t = threadIdx.x;
    for (int i = t; i < Ka; i += 256) ldsx[i] = xa[i];
    if (Kb > 0)
        for (int i = t; i < Kb; i += 256) ldsx[Ka + i] = xb[i];
    __syncthreads();

    // Wave-uniform problem selection; force into SGPRs so the K loop is a
    // plain scalar-counted loop (no exec-mask loop, minimal SALU overhead).
    const int wave = __builtin_amdgcn_readfirstlane(blockIdx.x * 8 + (t >> 5));
    if (wave >= tilesA + tilesB) return;
    const int selB  = __builtin_amdgcn_readfirstlane(wave >= tilesA ? 1 : 0);
    const int K     = __builtin_amdgcn_readfirstlane(selB ? Kb : Ka);
    const int tile  = __builtin_amdgcn_readfirstlane(selB ? (wave - tilesA) : wave);
    const int xoff  = selB ? Ka : 0;
    const float* W    = selB ? Wb : Wa;
    const float* bias = selB ? bb : ba;
    float*       out  = selB ? outb : outa;

    const int lane      = t & 31;
    const int half      = lane >> 4;        // 0: K+0/K+1 side, 1: K+2/K+3 side
    const int rowInTile = lane & 15;
    const int tb        = tile * 16;
    const v2f* w2 = (const v2f*)(W + (size_t)(tb + rowInTile) * (size_t)K);
    const float* xl = ldsx + xoff;
    const bool sel = (rowInTile == 0);      // lanes 0 and 16 carry column 0 of B

    v8f acc0 = {}, acc1 = {};
    for (int k0 = 0; k0 < K; k0 += 32) {
        v2f av[8];
#pragma unroll
        for (int u = 0; u < 8; ++u)         // 8 b64 NT loads issued as a clause
            av[u] = __builtin_nontemporal_load(w2 + ((k0 + 4 * u) >> 1) + half);
#pragma unroll
        for (int u = 0; u < 8; ++u) {
            const int kk = k0 + 4 * u + (half << 1);
            const float bx = xl[kk], by = xl[kk + 1];   // LDS broadcast
            v2f b; b[0] = sel ? bx : 0.f; b[1] = sel ? by : 0.f;
            if (u & 1) acc1 = wmma_f32_16x16x4(av[u], b, acc1);
            else       acc0 = wmma_f32_16x16x4(av[u], b, acc0);
        }
    }
    v8f acc = acc0 + acc1;

    if (sel) {                               // column 0 of D holds the matvec
        const int rb = tb + half * 8;
        for (int g = 0; g < 8; ++g) {
            float v = acc[g] + bias[rb + g];
            out[rb + g] = relu ? fmaxf(v, 0.f) : v;
        }
    }
}

// ---------------------------------------------------------------------------
// GRU gate combine: torch GRUCell gate order (r, z, n).
// ---------------------------------------------------------------------------
__global__ __launch_bounds__(256) void gru_combine(
    const float* __restrict__ gi, const float* __restrict__ gh,
    const float* __restrict__ h, float* __restrict__ out, int H)
{
    const int i = blockIdx.x * 256 + threadIdx.x;
    if (i >= H) return;
    const float r = 1.f / (1.f + expf(-(gi[i] + gh[i])));
    const float z = 1.f / (1.f + expf(-(gi[H + i] + gh[H + i])));
    const float n = tanhf(gi[2 * H + i] + r * gh[2 * H + i]);
    out[i] = (1.f - z) * n + z * h[i];
}

// ---------------------------------------------------------------------------
// FC2 layer 2: 1024 rows x 81920 cols. Block-per-row split-K reduction with
// float4 (b128) fully-coalesced NT streaming (335 MB read once); x (hidden,
// 320 KB) stays L2-resident via regular loads. K assumed multiple of 1024.
// ---------------------------------------------------------------------------
__global__ __launch_bounds__(256) void gemv_block_row(
    const float* __restrict__ W, const float* __restrict__ x,
    const float* __restrict__ b, float* __restrict__ out, int K)
{
    const int r = blockIdx.x, t = threadIdx.x;
    const v4f* w4 = (const v4f*)(W + (size_t)r * (size_t)K);
    const v4f* x4 = (const v4f*)x;
    const int n4 = K >> 2;
    const int iters = n4 >> 8;               // n4 / 256 (exact for K=81920)
    float acc = 0.f;
#pragma unroll 4
    for (int ii = 0; ii < iters; ++ii) {
        const int i = t + (ii << 8);
        v4f wv = __builtin_nontemporal_load(w4 + i);
        v4f xv = x4[i];
        acc += wv[0] * xv[0] + wv[1] * xv[1] + wv[2] * xv[2] + wv[3] * xv[3];
    }
    for (int i = t + (iters << 8); i < n4; i += 256) {   // remainder (none here)
        v4f wv = __builtin_nontemporal_load(w4 + i);
        v4f xv = x4[i];
        acc += wv[0] * xv[0] + wv[1] * xv[1] + wv[2] * xv[2] + wv[3] * xv[3];
    }
    __shared__ float red[256];
    red[t] = acc;
    __syncthreads();
    for (int s = 128; s > 0; s >>= 1) {
        if (t < s) red[t] += red[t + s];
        __syncthreads();
    }
    if (t == 0) out[r] = red[0] + b[r];
}

// ---------------------------------------------------------------------------
// Final: m1x_posterior_new = m1x_prior + KG @ dy   (KG = out_FC2 as 32x32)
// ---------------------------------------------------------------------------
__global__ void final_kernel(const float* __restrict__ ws, float* __restrict__ out)
{
    const int i = threadIdx.x;          // 32 threads
    const float* KG = ws + OFF_FC2OUT;
    const float* dy = ws + OFF_DY;
    float a = ws[OFF_PRIOR + i];
    for (int j = 0; j < 32; ++j) a += KG[i * 32 + j] * dy[j];
    out[i] = a;
}

// ---------------------------------------------------------------------------
extern "C" void kernel_launch(void* const* d_in, const int* in_sizes, int n_in,
                              void* d_out, int out_size, void* d_ws, size_t ws_size,
                              hipStream_t stream)
{
    (void)in_sizes; (void)n_in; (void)out_size; (void)ws_size;
    const float* y          = (const float*)d_in[0];
    const float* post       = (const float*)d_in[1];
    const float* post_prev  = (const float*)d_in[2];
    const float* prior_prev = (const float*)d_in[3];
    const float* y_prev     = (const float*)d_in[4];
    const float* h_Q        = (const float*)d_in[5];
    const float* h_Sigma    = (const float*)d_in[6];
    const float* h_S        = (const float*)d_in[7];
    const float* F          = (const float*)d_in[8];
    const float* H          = (const float*)d_in[9];
    const float* FC5_w      = (const float*)d_in[10];
    const float* FC5_b      = (const float*)d_in[11];
    const float* FC6_w      = (const float*)d_in[12];
    const float* FC6_b      = (const float*)d_in[13];
    const float* FC7_w      = (const float*)d_in[14];
    const float* FC7_b      = (const float*)d_in[15];
    const float* GRUQ_Wih   = (const float*)d_in[16];
    const float* GRUQ_Whh   = (const float*)d_in[17];
    const float* GRUQ_bih   = (const float*)d_in[18];
    const float* GRUQ_bhh   = (const float*)d_in[19];
    const float* GSig_Wih   = (const float*)d_in[20];
    const float* GSig_Whh   = (const float*)d_in[21];
    const float* GSig_bih   = (const float*)d_in[22];
    const float* GSig_bhh   = (const float*)d_in[23];
    const float* GS_Wih     = (const float*)d_in[24];
    const float* GS_Whh     = (const float*)d_in[25];
    const float* GS_bih     = (const float*)d_in[26];
    const float* GS_bhh     = (const float*)d_in[27];
    const float* FC1_w      = (const float*)d_in[28];
    const float* FC1_b      = (const float*)d_in[29];
    const float* FC2_w1     = (const float*)d_in[30];
    const float* FC2_b1     = (const float*)d_in[31];
    const float* FC2_w2     = (const float*)d_in[32];
    const float* FC2_b2     = (const float*)d_in[33];
    // d_in[34..37] = FC3/FC4 (dead code in reference), d_in[38] = fix_H_flag

    float* ws = (float*)d_ws;

    prep_kernel<<<1, 128, 0, stream>>>(y, post, post_prev, prior_prev, y_prev,
                                       F, H, FC5_w, FC5_b, FC6_w, FC6_b,
                                       FC7_w, FC7_b, ws);

    // GRU_Q: gi = Wih@out_FC5 + bih ; gh = Whh@h_Q + bhh  (384 tiles)
    wmma_gemv2<<<48, 256, 0, stream>>>(
        GRUQ_Wih, ws + OFF_FC5, GRUQ_bih, ws + OFF_GI, 160, 192,
        GRUQ_Whh, h_Q,          GRUQ_bhh, ws + OFF_GH, 1024, 192, 0);
    gru_combine<<<4, 256, 0, stream>>>(ws + OFF_GI, ws + OFF_GH, h_Q,
                                       ws + OFF_Q, 1024);

    // GRU_Sigma: x = concat(out_Q, out_FC6) contiguous at OFF_Q (len 1184)
    wmma_gemv2<<<48, 256, 0, stream>>>(
        GSig_Wih, ws + OFF_Q, GSig_bih, ws + OFF_GI, 1184, 192,
        GSig_Whh, h_Sigma,    GSig_bhh, ws + OFF_GH, 1024, 192, 0);
    gru_combine<<<4, 256, 0, stream>>>(ws + OFF_GI, ws + OFF_GH, h_Sigma,
                                       ws + OFF_SIGMA, 1024);

    // FC1: relu(FC1_w @ out_Sigma + b)  (64 tiles)
    wmma_gemv2<<<8, 256, 0, stream>>>(
        FC1_w, ws + OFF_SIGMA, FC1_b, ws + OFF_FC1, 1024, 64,
        nullptr, nullptr, nullptr, nullptr, 0, 0, 1);

    // GRU_S: x = concat(out_FC1, out_FC7) contiguous at OFF_FC1 (len 1344)
    wmma_gemv2<<<48, 256, 0, stream>>>(
        GS_Wih, ws + OFF_FC1, GS_bih, ws + OFF_GI, 1344, 192,
        GS_Whh, h_S,          GS_bhh, ws + OFF_GH, 1024, 192, 0);
    gru_combine<<<4, 256, 0, stream>>>(ws + OFF_GI, ws + OFF_GH, h_S,
                                       ws + OFF_S, 1024);

    // FC2 layer 1: relu(W1[81920x2048] @ concat(out_Sigma,out_S) + b1)
    // 5120 tiles / 8 waves per block = 640 blocks; streams 671 MB (dominant).
    wmma_gemv2<<<640, 256, 0, stream>>>(
        FC2_w1, ws + OFF_SIGMA, FC2_b1, ws + OFF_HIDDEN, 2048, 5120,
        nullptr, nullptr, nullptr, nullptr, 0, 0, 1);

    // FC2 layer 2: W2[1024x81920] @ hidden + b2  (block-per-row, 335 MB)
    gemv_block_row<<<1024, 256, 0, stream>>>(FC2_w2, ws + OFF_HIDDEN, FC2_b2,
                                             ws + OFF_FC2OUT, 81920);

    final_kernel<<<1, 32, 0, stream>>>(ws, (float*)d_out);
}